// GCNRegressor_53575422050753
// MI455X (gfx1250) — compile-verified
//
#include <hip/hip_runtime.h>
#include <hip/hip_bf16.h>

typedef __attribute__((ext_vector_type(2))) float v2f;
typedef __attribute__((ext_vector_type(8))) float v8f;

#define HID 64

// ---------------- utility ----------------
__global__ void k_zero(float* __restrict__ p, int n) {
    int i = blockIdx.x * blockDim.x + threadIdx.x;
    if (i < n) p[i] = 0.0f;
}

// ---------------- degrees ----------------
__global__ void k_deg(const int* __restrict__ src, const int* __restrict__ dst,
                      float* __restrict__ deg_out, float* __restrict__ deg_in, int E) {
    int e = blockIdx.x * blockDim.x + threadIdx.x;
    if (e < E) {
        atomicAdd(&deg_out[src[e]], 1.0f);
        atomicAdd(&deg_in[dst[e]], 1.0f);
    }
}

// norms + initial scalar feature h0 = in_deg
__global__ void k_norm(const float* __restrict__ deg_in, const float* __restrict__ deg_out,
                       float* __restrict__ norm_src, float* __restrict__ norm_dst,
                       float* __restrict__ h0, int N) {
    int i = blockIdx.x * blockDim.x + threadIdx.x;
    if (i < N) {
        float o = deg_out[i]; o = o > 1.0f ? o : 1.0f;
        float d = deg_in[i];  float dc = d > 1.0f ? d : 1.0f;
        norm_src[i] = rsqrtf(o);
        norm_dst[i] = rsqrtf(dc);
        h0[i] = d;
    }
}

// ---------------- layer 0 (scalar feature) ----------------
__global__ void k_scatter1(const int* __restrict__ src, const int* __restrict__ dst,
                           const float* __restrict__ h0, const float* __restrict__ norm_src,
                           float* __restrict__ agg0, int E) {
    int e = blockIdx.x * blockDim.x + threadIdx.x;
    if (e < E) {
        int s = src[e];
        atomicAdd(&agg0[dst[e]], h0[s] * norm_src[s]);
    }
}

// h[i][j] = relu(agg0[i]*norm_dst[i]*W0[j] + b0[j])   (outer product, W0 is 1x64)
__global__ void k_layer0(const float* __restrict__ agg0, const float* __restrict__ norm_dst,
                         const float* __restrict__ W0, const float* __restrict__ b0,
                         float* __restrict__ h, int N) {
    int idx = blockIdx.x * blockDim.x + threadIdx.x;
    int i = idx >> 6, j = idx & 63;
    if (i < N) {
        float v = agg0[i] * norm_dst[i] * W0[j] + b0[j];
        h[idx] = v > 0.0f ? v : 0.0f;
    }
}

// ---------------- 64-wide scatter (message passing) ----------------
// 32 lanes per edge; each lane moves 2 of the 64 features.
__global__ void k_scatter64(const int* __restrict__ src, const int* __restrict__ dst,
                            const float* __restrict__ h, const float* __restrict__ norm_src,
                            float* __restrict__ agg, int E) {
    int t = blockIdx.x * blockDim.x + threadIdx.x;
    int e = t >> 5;
    int lane = t & 31;
    if (e < E) {
        int s = src[e], d = dst[e];
        float sc = norm_src[s];
        float v0 = h[s * HID + lane] * sc;
        float v1 = h[s * HID + 32 + lane] * sc;
        atomicAdd(&agg[d * HID + lane], v0);
        atomicAdd(&agg[d * HID + 32 + lane], v1);
    }
}

// ---------------- WMMA GEMM: hout = relu((agg .* norm_dst) @ W + b) ----------------
// One wave computes a 16-row x 64-col tile using V_WMMA_F32_16X16X4_F32.
// A fragment (16x4 f32): lanes 0-15 hold K={k0,k0+1}, lanes 16-31 hold K={k0+2,k0+3}, row = lane&15.
// B fragment (4x16 f32): same K split across lane halves; col = lane&15.
// C/D (16x16 f32): VGPR i -> row i + 8*(lane>=16); col = lane&15.
__global__ __launch_bounds__(256)
void k_gemm_wmma(const float* __restrict__ agg, const float* __restrict__ norm_dst,
                 const float* __restrict__ W,    // 64x64 row-major [k][n]
                 const float* __restrict__ b,    // 64
                 float* __restrict__ hout,       // N x 64
                 int N) {
    const int wave = blockIdx.x * (blockDim.x >> 5) + (threadIdx.x >> 5);
    const int lane = threadIdx.x & 31;
    const int row0 = wave * 16;
    if (row0 >= N) return;                 // wave-uniform exit: EXEC stays all-ones for WMMA

    const int half = lane >> 4;            // 0: lanes 0-15, 1: lanes 16-31
    const int l16  = lane & 15;

    int arow = row0 + l16;
    if (arow > N - 1) arow = N - 1;        // clamp (no branch -> EXEC stays full)
    const float nd = norm_dst[arow];
    const float* arowp = agg + (size_t)arow * HID;

    v8f c0 = {}, c1 = {}, c2 = {}, c3 = {};

    #pragma unroll
    for (int k0 = 0; k0 < HID; k0 += 4) {
        const int kk = k0 + 2 * half;      // this lane-half's K pair
        v2f a;
        a.x = arowp[kk]     * nd;
        a.y = arowp[kk + 1] * nd;

        const float* w0p = W + (size_t)kk * HID;        // row K=kk
        const float* w1p = W + (size_t)(kk + 1) * HID;  // row K=kk+1
        v2f bf0, bf1, bf2, bf3;
        bf0.x = w0p[ 0 + l16]; bf0.y = w1p[ 0 + l16];
        bf1.x = w0p[16 + l16]; bf1.y = w1p[16 + l16];
        bf2.x = w0p[32 + l16]; bf2.y = w1p[32 + l16];
        bf3.x = w0p[48 + l16]; bf3.y = w1p[48 + l16];

        c0 = __builtin_amdgcn_wmma_f32_16x16x4_f32(false, a, false, bf0, (short)0, c0, false, false);
        c1 = __builtin_amdgcn_wmma_f32_16x16x4_f32(false, a, false, bf1, (short)0, c1, false, false);
        c2 = __builtin_amdgcn_wmma_f32_16x16x4_f32(false, a, false, bf2, (short)0, c2, false, false);
        c3 = __builtin_amdgcn_wmma_f32_16x16x4_f32(false, a, false, bf3, (short)0, c3, false, false);
    }

    // store: bias + relu
    #pragma unroll
    for (int i = 0; i < 8; ++i) {
        const int row = row0 + i + 8 * half;
        if (row < N) {
            float* rp = hout + (size_t)row * HID;
            float v;
            v = c0[i] + b[ 0 + l16]; rp[ 0 + l16] = v > 0.0f ? v : 0.0f;
            v = c1[i] + b[16 + l16]; rp[16 + l16] = v > 0.0f ? v : 0.0f;
            v = c2[i] + b[32 + l16]; rp[32 + l16] = v > 0.0f ? v : 0.0f;
            v = c3[i] + b[48 + l16]; rp[48 + l16] = v > 0.0f ? v : 0.0f;
        }
    }
}

// ---------------- readout ----------------
__global__ void k_count(const int* __restrict__ n2g, float* __restrict__ counts, int N) {
    int i = blockIdx.x * blockDim.x + threadIdx.x;
    if (i < N) atomicAdd(&counts[n2g[i]], 1.0f);
}

__global__ void k_pool(const int* __restrict__ n2g, const float* __restrict__ h,
                       float* __restrict__ hg, int N) {
    int t = blockIdx.x * blockDim.x + threadIdx.x;
    int i = t >> 5;
    int lane = t & 31;
    if (i < N) {
        int g = n2g[i];
        atomicAdd(&hg[g * HID + lane],      h[i * HID + lane]);
        atomicAdd(&hg[g * HID + 32 + lane], h[i * HID + 32 + lane]);
    }
}

__global__ void k_readout(const float* __restrict__ hg, const float* __restrict__ counts,
                          const float* __restrict__ Wr, const float* __restrict__ br,
                          float* __restrict__ out, int G) {
    int g = blockIdx.x * blockDim.x + threadIdx.x;
    if (g < G) {
        float cnt = counts[g]; cnt = cnt > 1.0f ? cnt : 1.0f;
        float acc = 0.0f;
        #pragma unroll
        for (int k = 0; k < HID; ++k) acc += hg[g * HID + k] * Wr[k];
        out[g] = acc / cnt + br[0];
    }
}

// ---------------- host orchestration ----------------
extern "C" void kernel_launch(void* const* d_in, const int* in_sizes, int n_in,
                              void* d_out, int out_size, void* d_ws, size_t ws_size,
                              hipStream_t stream) {
    const int*   src  = (const int*)d_in[0];
    const int*   dst  = (const int*)d_in[1];
    const int*   n2g  = (const int*)d_in[2];
    const float* W0   = (const float*)d_in[3];
    const float* b0   = (const float*)d_in[4];
    const float* W1   = (const float*)d_in[5];
    const float* b1   = (const float*)d_in[6];
    const float* W2   = (const float*)d_in[7];
    const float* b2   = (const float*)d_in[8];
    const float* Wr   = (const float*)d_in[9];
    const float* br   = (const float*)d_in[10];
    float* out = (float*)d_out;

    const int E = in_sizes[0];
    const int N = in_sizes[2];
    const int G = out_size;

    // workspace layout (floats)
    float* ws       = (float*)d_ws;
    float* deg_in   = ws;                    // N
    float* deg_out  = deg_in  + N;           // N
    float* norm_src = deg_out + N;           // N
    float* norm_dst = norm_src + N;          // N
    float* agg0     = norm_dst + N;          // N
    float* h        = agg0 + N;              // N*64
    float* agg      = h + (size_t)N * HID;   // N*64
    float* counts   = agg + (size_t)N * HID; // G
    float* hg       = counts + G;            // G*64

    const int TB = 256;
    auto blocks = [&](long n) { return (unsigned)((n + TB - 1) / TB); };

    // degrees
    k_zero<<<blocks(2L * N), TB, 0, stream>>>(deg_in, 2 * N);   // deg_in, deg_out contiguous
    k_deg<<<blocks(E), TB, 0, stream>>>(src, dst, deg_out, deg_in, E);
    k_norm<<<blocks(N), TB, 0, stream>>>(deg_in, deg_out, norm_src, norm_dst, agg0 /*h0 scratch*/, N);

    // layer 0 (scalar): reuse deg_in as agg0 target? keep explicit: zero a fresh scalar acc
    // h0 lives in agg0; accumulate into deg_in slot reused as scalar agg
    k_zero<<<blocks(N), TB, 0, stream>>>(deg_in, N);            // deg_in reused as scalar agg
    k_scatter1<<<blocks(E), TB, 0, stream>>>(src, dst, agg0, norm_src, deg_in, E);
    k_layer0<<<blocks((long)N * HID), TB, 0, stream>>>(deg_in, norm_dst, W0, b0, h, N);

    const unsigned gemm_blocks = (unsigned)(((N + 15) / 16 + (TB / 32) - 1) / (TB / 32));

    // layer 1
    k_zero<<<blocks((long)N * HID), TB, 0, stream>>>(agg, N * HID);
    k_scatter64<<<blocks((long)E * 32), TB, 0, stream>>>(src, dst, h, norm_src, agg, E);
    k_gemm_wmma<<<gemm_blocks, TB, 0, stream>>>(agg, norm_dst, W1, b1, h, N);

    // layer 2
    k_zero<<<blocks((long)N * HID), TB, 0, stream>>>(agg, N * HID);
    k_scatter64<<<blocks((long)E * 32), TB, 0, stream>>>(src, dst, h, norm_src, agg, E);
    k_gemm_wmma<<<gemm_blocks, TB, 0, stream>>>(agg, norm_dst, W2, b2, h, N);

    // readout
    k_zero<<<blocks(G + (long)G * HID), TB, 0, stream>>>(counts, G + G * HID); // counts+hg contiguous
    k_count<<<blocks(N), TB, 0, stream>>>(n2g, counts, N);
    k_pool<<<blocks((long)N * 32), TB, 0, stream>>>(n2g, h, hg, N);
    k_readout<<<1, 64, 0, stream>>>(hg, counts, Wr, br, out, G);
}